// GNNWithGlobal_34437047779687
// MI455X (gfx1250) — compile-verified
//
#include <hip/hip_runtime.h>
#include <math.h>

typedef float v2f __attribute__((ext_vector_type(2)));
typedef float v8f __attribute__((ext_vector_type(8)));

#define F 128          // feature width (NODE_IN == HIDDEN == 128)
#define GIN 64         // global feature width

__device__ __forceinline__ void atomAddF(float* p, float v) {
    unsafeAtomicAdd(p, v);
}

// ---------------------------------------------------------------- utilities
__global__ void k_zero(float* p, int n) {
    int i = blockIdx.x * blockDim.x + threadIdx.x;
    if (i < n) p[i] = 0.0f;
}

__global__ void k_fill(float* p, float v, int n) {
    int i = blockIdx.x * blockDim.x + threadIdx.x;
    if (i < n) p[i] = v;
}

// ew[e] = exp(-(edge_attr[e]/5)^2)
__global__ void k_edge_w(const float* __restrict__ ea, float* __restrict__ ew, int E) {
    int i = blockIdx.x * blockDim.x + threadIdx.x;
    if (i < E) {
        float t = ea[i] * 0.2f;
        ew[i] = expf(-t * t);
    }
}

// deg[dst[e]] += ew[e]   (deg pre-filled with 1.0 for self loops)
__global__ void k_deg(const float* __restrict__ ew, const int* __restrict__ dst,
                      float* __restrict__ deg, int E) {
    int i = blockIdx.x * blockDim.x + threadIdx.x;
    if (i < E) atomAddF(&deg[dst[i]], ew[i]);
}

// in-place deg -> dinv = rsqrt(deg) (deg > 0 -> rsqrt else 0)
__global__ void k_dinv(float* __restrict__ deg, int N) {
    int i = blockIdx.x * blockDim.x + threadIdx.x;
    if (i < N) {
        float d = deg[i];
        deg[i] = (d > 0.0f) ? rsqrtf(d) : 0.0f;
    }
}

// ---------------------------------------------------------------- WMMA GEMM
// C[M x 128] = A[M x 128] @ B[128 x 128], fp32, V_WMMA_F32_16X16X4_F32.
// One wave computes a 16-row x 128-col stripe; M must be a multiple of 16.
__global__ void k_gemm_wmma(const float* __restrict__ A, const float* __restrict__ B,
                            float* __restrict__ C, int M) {
    int wavesPerBlock = blockDim.x >> 5;
    int wave = blockIdx.x * wavesPerBlock + (threadIdx.x >> 5);
    int lane = threadIdx.x & 31;
    int row0 = wave * 16;
    if (row0 >= M) return;                 // whole-wave uniform exit: EXEC stays all-1s

    int lg = lane >> 4;                    // 0: lanes 0-15, 1: lanes 16-31
    int lm = lane & 15;

    v8f acc[8] = {};                       // 8 N-tiles of 16 cols each

    const float* arow = A + (size_t)(row0 + lm) * F;
    for (int k = 0; k < F; k += 4) {
        // A fragment 16x4: lane-half selects K pair (ISA 7.12.2, 32-bit A 16x4)
        v2f a;
        a.x = arow[k + 2 * lg + 0];
        a.y = arow[k + 2 * lg + 1];
#pragma unroll
        for (int n = 0; n < 8; ++n) {
            // B fragment 4x16: VGPR r holds K = 2*lg + r, column = lm
            v2f b;
            b.x = B[(size_t)(k + 2 * lg + 0) * F + n * 16 + lm];
            b.y = B[(size_t)(k + 2 * lg + 1) * F + n * 16 + lm];
            acc[n] = __builtin_amdgcn_wmma_f32_16x16x4_f32(
                false, a, false, b, (short)0, acc[n], false, false);
        }
    }
    // C/D layout: VGPR r -> row r + 8*lg, col = lm  (16x16 f32 C/D table)
#pragma unroll
    for (int n = 0; n < 8; ++n) {
#pragma unroll
        for (int r = 0; r < 8; ++r) {
            int row = row0 + r + 8 * lg;
            C[(size_t)row * F + n * 16 + lm] = acc[n][r];
        }
    }
}

// ---------------------------------------------------------------- GCN aggregation
// Self-loop init: out[i,:] = xw[i,:] * dinv[i]^2   (one thread per float4)
__global__ void k_selfloop(const float* __restrict__ xw, const float* __restrict__ dinv,
                           float* __restrict__ out, int N) {
    int i = blockIdx.x * blockDim.x + threadIdx.x;   // over N * (F/4)
    if (i >= N * (F / 4)) return;
    int node = i >> 5;                               // F/4 == 32 chunks per node
    float s = dinv[node];
    s = s * s;
    float4 v = ((const float4*)xw)[i];
    v.x *= s; v.y *= s; v.z *= s; v.w *= s;
    ((float4*)out)[i] = v;
}

// One wave per edge: out[dst,:] += xw[src,:] * (dinv[src]*ew*dinv[dst])
__global__ void k_edge_scatter(const float* __restrict__ xw,
                               const int* __restrict__ srcI, const int* __restrict__ dstI,
                               const float* __restrict__ ew, const float* __restrict__ dinv,
                               float* __restrict__ out, int E) {
    int wavesPerBlock = blockDim.x >> 5;
    int e = blockIdx.x * wavesPerBlock + (threadIdx.x >> 5);
    if (e >= E) return;
    int lane = threadIdx.x & 31;
    int s = srcI[e], d = dstI[e];
    float norm = dinv[s] * ew[e] * dinv[d];
    float4 v = ((const float4*)(xw + (size_t)s * F))[lane];
    float* o = out + (size_t)d * F + lane * 4;
    atomAddF(o + 0, v.x * norm);
    atomAddF(o + 1, v.y * norm);
    atomAddF(o + 2, v.z * norm);
    atomAddF(o + 3, v.w * norm);
}

// h = relu(h + bias[col])
__global__ void k_bias_relu(float* __restrict__ h, const float* __restrict__ b, int total) {
    int i = blockIdx.x * blockDim.x + threadIdx.x;
    if (i < total) {
        float v = h[i] + b[i & (F - 1)];
        h[i] = fmaxf(v, 0.0f);
    }
}

// ---------------------------------------------------------------- pooling + head
// One wave per node: sums[batch[n],:] += h[n,:]; counts[batch[n]] += 1
__global__ void k_pool(const float* __restrict__ h, const int* __restrict__ batch,
                       float* __restrict__ sums, float* __restrict__ counts, int N) {
    int wavesPerBlock = blockDim.x >> 5;
    int node = blockIdx.x * wavesPerBlock + (threadIdx.x >> 5);
    if (node >= N) return;
    int lane = threadIdx.x & 31;
    int g = batch[node];
    float4 v = ((const float4*)(h + (size_t)node * F))[lane];
    float* s = sums + (size_t)g * F + lane * 4;
    atomAddF(s + 0, v.x);
    atomAddF(s + 1, v.y);
    atomAddF(s + 2, v.z);
    atomAddF(s + 3, v.w);
    if (lane == 0) atomAddF(&counts[g], 1.0f);
}

// Per-graph head: out[g] = relu([mean | u] @ W3 + b3) @ W4 + b4
__global__ void k_head(const float* __restrict__ sums, const float* __restrict__ counts,
                       const float* __restrict__ u,
                       const float* __restrict__ W3, const float* __restrict__ b3,
                       const float* __restrict__ W4, const float* __restrict__ b4,
                       float* __restrict__ out) {
    __shared__ float gu[F + GIN];
    __shared__ float red[F];
    int g = blockIdx.x;
    int t = threadIdx.x;                       // blockDim.x == 128
    float cnt = fmaxf(counts[g], 1.0f);
    gu[t] = sums[(size_t)g * F + t] / cnt;
    if (t < GIN) gu[F + t] = u[(size_t)g * GIN + t];
    __syncthreads();
    float acc = b3[t];
    for (int j = 0; j < F + GIN; ++j)
        acc = fmaf(gu[j], W3[(size_t)j * F + t], acc);
    red[t] = fmaxf(acc, 0.0f) * W4[t];
    __syncthreads();
    for (int s = F / 2; s > 0; s >>= 1) {
        if (t < s) red[t] += red[t + s];
        __syncthreads();
    }
    if (t == 0) out[g] = red[0] + b4[0];
}

// ---------------------------------------------------------------- launcher
extern "C" void kernel_launch(void* const* d_in, const int* in_sizes, int n_in,
                              void* d_out, int out_size, void* d_ws, size_t ws_size,
                              hipStream_t stream) {
    const float* x   = (const float*)d_in[0];
    const int*   ei  = (const int*)d_in[1];
    const float* ea  = (const float*)d_in[2];
    const int*   bat = (const int*)d_in[3];
    const float* u   = (const float*)d_in[4];
    const float* W1  = (const float*)d_in[5];
    const float* b1  = (const float*)d_in[6];
    const float* W2  = (const float*)d_in[7];
    const float* b2  = (const float*)d_in[8];
    const float* W3  = (const float*)d_in[9];
    const float* b3  = (const float*)d_in[10];
    const float* W4  = (const float*)d_in[11];
    const float* b4  = (const float*)d_in[12];
    float* out = (float*)d_out;

    const int N = in_sizes[0] / F;          // 100000
    const int E = in_sizes[2];              // 1600000
    const int G = in_sizes[4] / GIN;        // 512

    const int* srcI = ei;
    const int* dstI = ei + E;

    // workspace carving (floats)
    float* ws = (float*)d_ws;
    float* ew     = ws;                          // E
    float* dinv   = ew + E;                      // N   (deg -> dinv in place)
    float* bufA   = dinv + N;                    // N*F (xw / hw)
    float* bufB   = bufA + (size_t)N * F;        // N*F (h1)
    float* bufC   = bufB + (size_t)N * F;        // N*F (h2)
    float* sums   = bufC + (size_t)N * F;        // G*F
    float* counts = sums + (size_t)G * F;        // G

    const int T = 256;
    const int wavesPerBlock = T / 32;
    int gridE   = (E + T - 1) / T;
    int gridN   = (N + T - 1) / T;
    int gridNF4 = (N * (F / 4) + T - 1) / T;
    int gridNF  = (N * F + T - 1) / T;
    int gridEw  = (E + wavesPerBlock - 1) / wavesPerBlock;   // wave-per-edge
    int gridNw  = (N + wavesPerBlock - 1) / wavesPerBlock;   // wave-per-node
    int mTiles  = N / 16;                                    // 6250 (N % 16 == 0)
    int gridGemm = (mTiles + 4 - 1) / 4;                     // 4 waves / 128-thr block

    // edge weights + normalization
    k_edge_w<<<gridE, T, 0, stream>>>(ea, ew, E);
    k_fill  <<<gridN, T, 0, stream>>>(dinv, 1.0f, N);        // self-loop weight
    k_deg   <<<gridE, T, 0, stream>>>(ew, dstI, dinv, E);
    k_dinv  <<<gridN, T, 0, stream>>>(dinv, N);
    k_zero  <<<(G * F + G + T - 1) / T, T, 0, stream>>>(sums, G * F + G);

    // layer 1: xw = x @ W1 ; aggregate ; relu(+b1)
    k_gemm_wmma   <<<gridGemm, 128, 0, stream>>>(x, W1, bufA, N);
    k_selfloop    <<<gridNF4, T, 0, stream>>>(bufA, dinv, bufB, N);
    k_edge_scatter<<<gridEw, T, 0, stream>>>(bufA, srcI, dstI, ew, dinv, bufB, E);
    k_bias_relu   <<<gridNF, T, 0, stream>>>(bufB, b1, N * F);

    // layer 2: hw = h1 @ W2 ; aggregate ; relu(+b2)
    k_gemm_wmma   <<<gridGemm, 128, 0, stream>>>(bufB, W2, bufA, N);
    k_selfloop    <<<gridNF4, T, 0, stream>>>(bufA, dinv, bufC, N);
    k_edge_scatter<<<gridEw, T, 0, stream>>>(bufA, srcI, dstI, ew, dinv, bufC, E);
    k_bias_relu   <<<gridNF, T, 0, stream>>>(bufC, b2, N * F);

    // pool + head
    k_pool<<<gridNw, T, 0, stream>>>(bufC, bat, sums, counts, N);
    k_head<<<G, F, 0, stream>>>(sums, counts, u, W3, b3, W4, b4, out);
}